// RNNPPAgent_49469433315484
// MI455X (gfx1250) — compile-verified
//
#include <hip/hip_runtime.h>
#include <math.h>

// ---- CDNA5 WMMA types ----
typedef float v2f __attribute__((ext_vector_type(2)));
typedef float v8f __attribute__((ext_vector_type(8)));

#define TILE   16
#define HDIM   128
#define INDIM  204
#define NMECH  4
#define ADIM   14
#define KR     32      // router width
#define HS     129     // LDS row stride for 128-wide buffers (conflict-free)
#define RS     33      // LDS row stride for 32-wide buffers

__device__ __forceinline__ v8f wmma4(v2f a, v2f b, v8f c) {
  // D = A(16x4,f32) * B(4x16,f32) + C(16x16,f32)
  return __builtin_amdgcn_wmma_f32_16x16x4_f32(false, a, false, b, (short)0, c,
                                               false, false);
}

__device__ __forceinline__ float sigm(float x) { return 1.0f / (1.0f + expf(-x)); }

__global__ __launch_bounds__(32)
void rnnpp_agent_kernel(const float* __restrict__ X,      // [32768,204]
                        const float* __restrict__ HIN,    // [32768,128]
                        const float* __restrict__ FC1W,   // [4,204,128]
                        const float* __restrict__ FC1B,   // [4,128]
                        const float* __restrict__ WIH,    // [4,128,384]
                        const float* __restrict__ WHH,    // [4,128,384]
                        const float* __restrict__ BIH,    // [4,384]
                        const float* __restrict__ BHH,    // [4,384]
                        const float* __restrict__ FC2W,   // [4,128,14]
                        const float* __restrict__ FC2B,   // [4,14]
                        const float* __restrict__ QPW1,   // [128,32]
                        const float* __restrict__ QPB1,   // [32]
                        const float* __restrict__ QPW2,   // [32,32]
                        const float* __restrict__ QPB2,   // [32]
                        const float* __restrict__ QPW3,   // [32,32]
                        const float* __restrict__ QPB3,   // [32]
                        const float* __restrict__ KEYS,   // [4,1,32]
                        const float* __restrict__ WQ,     // [32,32]
                        const float* __restrict__ BQ,     // [32]
                        const float* __restrict__ WK,     // [32,32]
                        const float* __restrict__ BK,     // [32]
                        const float* __restrict__ UNZ,    // [32768,4]
                        float* __restrict__ OUTQ,         // [32768,14]
                        float* __restrict__ OUTH)         // [32768,128]
{
  __shared__ float hs[TILE * HS];     // hidden tile (row-major)
  __shared__ float xa[TILE * HS];     // relu(fc1) activations for current mech
  __shared__ float hn[TILE * HS];     // h_new for current mech
  __shared__ float rtA[TILE * RS];    // router ping
  __shared__ float rtB[TILE * RS];    // router pong
  __shared__ float kp[NMECH * KR];    // projected mechanism keys
  __shared__ int   argm[TILE];        // selected mechanism per row

  const int lane = threadIdx.x;       // wave32
  const int lo   = lane & 15;
  const int hi   = lane >> 4;         // 0 | 1  (K-half selector for f32 frags)
  const int tok0 = blockIdx.x * TILE;

  // weights are L2-resident: hint them in early
  __builtin_prefetch(WIH, 0, 1);
  __builtin_prefetch(WHH, 0, 1);
  __builtin_prefetch(FC1W, 0, 1);

  // ---- stage hidden tile into LDS (coalesced) ----
  for (int i = lane; i < TILE * HDIM; i += 32) {
    int r = i >> 7, c = i & 127;
    hs[r * HS + c] = HIN[(tok0 + r) * HDIM + c];
  }
  __syncthreads();  // single-wave WG: NOP barrier, keeps DS ordering explicit

  // =========================================================================
  // Router (depends only on hidden_state): query MLP -> qp; kp; argmax select
  // =========================================================================
  // layer1: relu(hs[16,128] @ QPW1[128,32] + b1) -> rtA
  for (int nt = 0; nt < 2; ++nt) {
    const int col = nt * 16 + lo;
    v8f acc; { float b0 = QPB1[col]; for (int e = 0; e < 8; ++e) acc[e] = b0; }
    for (int ks = 0; ks < 32; ++ks) {
      const int kb = (ks << 2) + (hi << 1);
      v2f a; a.x = hs[lo * HS + kb];       a.y = hs[lo * HS + kb + 1];
      v2f b; b.x = QPW1[kb * KR + col];    b.y = QPW1[(kb + 1) * KR + col];
      acc = wmma4(a, b, acc);
    }
    for (int e = 0; e < 8; ++e) { int r = e + (hi << 3); rtA[r * RS + col] = fmaxf(acc[e], 0.f); }
  }
  // layer2: relu(rtA @ QPW2 + b2) -> rtB
  for (int nt = 0; nt < 2; ++nt) {
    const int col = nt * 16 + lo;
    v8f acc; { float b0 = QPB2[col]; for (int e = 0; e < 8; ++e) acc[e] = b0; }
    for (int ks = 0; ks < 8; ++ks) {
      const int kb = (ks << 2) + (hi << 1);
      v2f a; a.x = rtA[lo * RS + kb];      a.y = rtA[lo * RS + kb + 1];
      v2f b; b.x = QPW2[kb * KR + col];    b.y = QPW2[(kb + 1) * KR + col];
      acc = wmma4(a, b, acc);
    }
    for (int e = 0; e < 8; ++e) { int r = e + (hi << 3); rtB[r * RS + col] = fmaxf(acc[e], 0.f); }
  }
  // layer3: query = rtB @ QPW3 + b3 -> rtA
  for (int nt = 0; nt < 2; ++nt) {
    const int col = nt * 16 + lo;
    v8f acc; { float b0 = QPB3[col]; for (int e = 0; e < 8; ++e) acc[e] = b0; }
    for (int ks = 0; ks < 8; ++ks) {
      const int kb = (ks << 2) + (hi << 1);
      v2f a; a.x = rtB[lo * RS + kb];      a.y = rtB[lo * RS + kb + 1];
      v2f b; b.x = QPW3[kb * KR + col];    b.y = QPW3[(kb + 1) * KR + col];
      acc = wmma4(a, b, acc);
    }
    for (int e = 0; e < 8; ++e) { int r = e + (hi << 3); rtA[r * RS + col] = acc[e]; }
  }
  // qp = query @ WQ + BQ -> rtB
  for (int nt = 0; nt < 2; ++nt) {
    const int col = nt * 16 + lo;
    v8f acc; { float b0 = BQ[col]; for (int e = 0; e < 8; ++e) acc[e] = b0; }
    for (int ks = 0; ks < 8; ++ks) {
      const int kb = (ks << 2) + (hi << 1);
      v2f a; a.x = rtA[lo * RS + kb];      a.y = rtA[lo * RS + kb + 1];
      v2f b; b.x = WQ[kb * KR + col];      b.y = WQ[(kb + 1) * KR + col];
      acc = wmma4(a, b, acc);
    }
    for (int e = 0; e < 8; ++e) { int r = e + (hi << 3); rtB[r * RS + col] = acc[e]; }
  }
  // kp[m] = keys[m] @ WK + BK   (tiny: each lane owns one output column)
  for (int m = 0; m < NMECH; ++m) {
    float s = BK[lane];
    for (int k = 0; k < KR; ++k) s += KEYS[m * KR + k] * WK[k * KR + lane];
    kp[m * KR + lane] = s;
  }
  // softmax(attn) + gumbel -> hard argmax (score == y_hard in forward pass)
  if (lane < TILE) {
    const int r = lane, tok = tok0 + r;
    const float scale = 0.17677669529663687f;   // 1/sqrt(32)
    float l[NMECH];
    for (int m = 0; m < NMECH; ++m) {
      float s = 0.f;
      for (int k = 0; k < KR; ++k) s += rtB[r * RS + k] * kp[m * KR + k];
      l[m] = s * scale;
    }
    float mx = fmaxf(fmaxf(l[0], l[1]), fmaxf(l[2], l[3]));
    float ex[NMECH], sum = 0.f;
    for (int m = 0; m < NMECH; ++m) { ex[m] = expf(l[m] - mx); sum += ex[m]; }
    float best = -1e30f; int bi = 0;
    for (int m = 0; m < NMECH; ++m) {
      float u = UNZ[tok * NMECH + m];
      float g = -logf(-logf(u + 1e-10f) + 1e-10f);
      float t = ex[m] / sum + g;                // attn + gumbel, tau = 1
      if (t > best) { best = t; bi = m; }
    }
    argm[r] = bi;
  }
  __syncthreads();

  // =========================================================================
  // Per-mechanism RNNAgent; only rows with argm==m are written out
  // =========================================================================
  for (int m = 0; m < NMECH; ++m) {
    // ---- fc1: xa = relu(x @ FC1W[m] + FC1B[m]), K = 204 -> 51 wmma steps ----
    const float* fw = FC1W + m * INDIM * HDIM;
    for (int nt = 0; nt < 8; ++nt) {
      const int col = nt * 16 + lo;
      v8f acc; { float b0 = FC1B[m * HDIM + col]; for (int e = 0; e < 8; ++e) acc[e] = b0; }
      const float* xr = X + (size_t)(tok0 + lo) * INDIM;
      for (int ks = 0; ks < 51; ++ks) {
        const int kb = (ks << 2) + (hi << 1);
        v2f a; a.x = xr[kb];                a.y = xr[kb + 1];
        v2f b; b.x = fw[kb * HDIM + col];   b.y = fw[(kb + 1) * HDIM + col];
        acc = wmma4(a, b, acc);
      }
      for (int e = 0; e < 8; ++e) { int r = e + (hi << 3); xa[r * HS + col] = fmaxf(acc[e], 0.f); }
    }
    __syncthreads();

    // ---- fused GRU gates: gi+gh for r,z in one chain; n-gate split ----
    const float* wih = WIH + m * HDIM * 384;
    const float* whh = WHH + m * HDIM * 384;
    for (int nt = 0; nt < 8; ++nt) {
      const int col = nt * 16 + lo;
      v8f ar, az, ani, anh;
      { float br  = BIH[m * 384 + col]       + BHH[m * 384 + col];
        float bz  = BIH[m * 384 + 128 + col] + BHH[m * 384 + 128 + col];
        float bni = BIH[m * 384 + 256 + col];
        float bnh = BHH[m * 384 + 256 + col];
        for (int e = 0; e < 8; ++e) { ar[e] = br; az[e] = bz; ani[e] = bni; anh[e] = bnh; } }
      for (int ks = 0; ks < 32; ++ks) {
        const int kb = (ks << 2) + (hi << 1);
        v2f ax; ax.x = xa[lo * HS + kb]; ax.y = xa[lo * HS + kb + 1];
        v2f ah; ah.x = hs[lo * HS + kb]; ah.y = hs[lo * HS + kb + 1];
        v2f b;
        b.x = wih[kb * 384 + col];        b.y = wih[(kb + 1) * 384 + col];        ar  = wmma4(ax, b, ar);
        b.x = whh[kb * 384 + col];        b.y = whh[(kb + 1) * 384 + col];        ar  = wmma4(ah, b, ar);
        b.x = wih[kb * 384 + 128 + col];  b.y = wih[(kb + 1) * 384 + 128 + col];  az  = wmma4(ax, b, az);
        b.x = whh[kb * 384 + 128 + col];  b.y = whh[(kb + 1) * 384 + 128 + col];  az  = wmma4(ah, b, az);
        b.x = wih[kb * 384 + 256 + col];  b.y = wih[(kb + 1) * 384 + 256 + col];  ani = wmma4(ax, b, ani);
        b.x = whh[kb * 384 + 256 + col];  b.y = whh[(kb + 1) * 384 + 256 + col];  anh = wmma4(ah, b, anh);
      }
      for (int e = 0; e < 8; ++e) {
        const int r = e + (hi << 3);
        float rg = sigm(ar[e]);
        float zg = sigm(az[e]);
        float ng = tanhf(ani[e] + rg * anh[e]);
        float h0 = hs[r * HS + col];
        hn[r * HS + col] = (1.f - zg) * ng + zg * h0;
      }
    }
    __syncthreads();

    // ---- fc2: q = h_new @ FC2W[m] + FC2B[m]  (A=14, pad N-tile to 16) ----
    {
      const int col = lo;
      const bool ok = (col < ADIM);
      v8f q; { float b0 = ok ? FC2B[m * ADIM + col] : 0.f; for (int e = 0; e < 8; ++e) q[e] = b0; }
      const float* f2 = FC2W + m * HDIM * ADIM;
      for (int ks = 0; ks < 32; ++ks) {
        const int kb = (ks << 2) + (hi << 1);
        v2f a; a.x = hn[lo * HS + kb]; a.y = hn[lo * HS + kb + 1];
        v2f b; b.x = ok ? f2[kb * ADIM + col] : 0.f;
               b.y = ok ? f2[(kb + 1) * ADIM + col] : 0.f;
        q = wmma4(a, b, q);
      }
      for (int e = 0; e < 8; ++e) {
        const int r = e + (hi << 3);
        if (argm[r] == m && ok) OUTQ[(size_t)(tok0 + r) * ADIM + col] = q[e];
      }
    }
    // ---- write selected h_new rows (argm uniform across wave) ----
    for (int r = 0; r < TILE; ++r) {
      if (argm[r] == m) {
        for (int c = lane; c < HDIM; c += 32)
          OUTH[(size_t)(tok0 + r) * HDIM + c] = hn[r * HS + c];
      }
    }
    __syncthreads();
  }
}

extern "C" void kernel_launch(void* const* d_in, const int* in_sizes, int n_in,
                              void* d_out, int out_size, void* d_ws, size_t ws_size,
                              hipStream_t stream) {
  (void)n_in; (void)out_size; (void)d_ws; (void)ws_size;
  const float* X    = (const float*)d_in[0];
  const float* HIN  = (const float*)d_in[1];
  const float* FC1W = (const float*)d_in[2];
  const float* FC1B = (const float*)d_in[3];
  const float* WIH  = (const float*)d_in[4];
  const float* WHH  = (const float*)d_in[5];
  const float* BIH  = (const float*)d_in[6];
  const float* BHH  = (const float*)d_in[7];
  const float* FC2W = (const float*)d_in[8];
  const float* FC2B = (const float*)d_in[9];
  const float* QPW1 = (const float*)d_in[10];
  const float* QPB1 = (const float*)d_in[11];
  const float* QPW2 = (const float*)d_in[12];
  const float* QPB2 = (const float*)d_in[13];
  const float* QPW3 = (const float*)d_in[14];
  const float* QPB3 = (const float*)d_in[15];
  const float* KEYS = (const float*)d_in[16];
  const float* WQ   = (const float*)d_in[17];
  const float* BQ   = (const float*)d_in[18];
  const float* WK   = (const float*)d_in[19];
  const float* BK   = (const float*)d_in[20];
  const float* UNZ  = (const float*)d_in[21];

  const int tokens = in_sizes[1] / HDIM;       // 32768
  float* OUTQ = (float*)d_out;                 // [tokens,14]
  float* OUTH = (float*)d_out + (size_t)tokens * ADIM;  // [tokens,128]

  dim3 grid(tokens / TILE), block(32);
  rnnpp_agent_kernel<<<grid, block, 0, stream>>>(
      X, HIN, FC1W, FC1B, WIH, WHH, BIH, BHH, FC2W, FC2B,
      QPW1, QPB1, QPW2, QPB2, QPW3, QPB3, KEYS, WQ, BQ, WK, BK, UNZ,
      OUTQ, OUTH);
}